// CraphCNN_2989297238480
// MI455X (gfx1250) — compile-verified
//
#include <hip/hip_runtime.h>
#include <math.h>

typedef __attribute__((ext_vector_type(2))) float v2f;
typedef __attribute__((ext_vector_type(8))) float v8f;

#define IN_DIM  128
#define HID_DIM 64
#define OUT_DIM 16

// ---------------- degree / normalization ----------------
__global__ void k_deg_init(float* __restrict__ deg, int n) {
    int i = blockIdx.x * blockDim.x + threadIdx.x;
    if (i < n) deg[i] = 1.0f;                       // self-loop contributes 1
}

__global__ void k_deg_count(const int* __restrict__ dst, float* __restrict__ deg, int e) {
    int i = blockIdx.x * blockDim.x + threadIdx.x;
    if (i < e) atomicAdd(&deg[dst[i]], 1.0f);
}

__global__ void k_dinv(float* __restrict__ deg, int n) {
    int i = blockIdx.x * blockDim.x + threadIdx.x;
    if (i < n) deg[i] = rsqrtf(deg[i]);             // deg >= 1 always
}

// ---------------- GEMM1: h1 = x @ W1 ; agg1 = h1 * dinv^2 (self loop) ----------------
// One wave computes a 16x64 output stripe (4 WMMA col-tiles, A reused across tiles).
__global__ void k_gemm1(const float* __restrict__ x, const float* __restrict__ W1,
                        const float* __restrict__ dinv,
                        float* __restrict__ h1, float* __restrict__ agg1, int n) {
    const int lane  = threadIdx.x & 31;
    const int wave  = threadIdx.x >> 5;
    const int rtile = blockIdx.x * 8 + wave;
    const int rbase = rtile * 16;
    if (rbase >= n) return;                         // wave-uniform exit: EXEC stays full

    const int half = lane >> 4;                     // 0: K lo pair, 1: K hi pair
    const int mrow = lane & 15;
    int arow = rbase + mrow;
    if (arow > n - 1) arow = n - 1;                 // branchless clamp, EXEC stays full

    v8f acc[4];
#pragma unroll
    for (int t = 0; t < 4; ++t)
#pragma unroll
        for (int q = 0; q < 8; ++q) acc[t][q] = 0.0f;

    for (int k0 = 0; k0 < IN_DIM; k0 += 4) {
        const int ka = k0 + 2 * half;
        v2f a;
        a.x = x[(long long)arow * IN_DIM + ka];
        a.y = x[(long long)arow * IN_DIM + ka + 1];
#pragma unroll
        for (int t = 0; t < 4; ++t) {
            const int nb = t * 16 + mrow;
            v2f b;
            b.x = W1[ka * HID_DIM + nb];
            b.y = W1[(ka + 1) * HID_DIM + nb];
            acc[t] = __builtin_amdgcn_wmma_f32_16x16x4_f32(
                false, a, false, b, (short)0, acc[t], false, false);
        }
    }

    if (rbase + 16 <= n) {
        // full tile: wave-uniform, no per-lane guards -> clauseable stores
#pragma unroll
        for (int g = 0; g < 8; ++g) {
            const int row = rbase + g + 8 * half;
            const float dv = dinv[row];
            const float sl = dv * dv;
#pragma unroll
            for (int t = 0; t < 4; ++t) {
                const int col = t * 16 + mrow;
                const float v = acc[t][g];
                h1[(long long)row * HID_DIM + col]   = v;
                agg1[(long long)row * HID_DIM + col] = v * sl;   // self-loop init
            }
        }
    } else {
#pragma unroll
        for (int g = 0; g < 8; ++g) {
            const int row = rbase + g + 8 * half;
            if (row < n) {
                const float dv = dinv[row];
                const float sl = dv * dv;
#pragma unroll
                for (int t = 0; t < 4; ++t) {
                    const int col = t * 16 + mrow;
                    const float v = acc[t][g];
                    h1[(long long)row * HID_DIM + col]   = v;
                    agg1[(long long)row * HID_DIM + col] = v * sl;
                }
            }
        }
    }
}

// ---------------- scatter layer 1: agg1[d] += h1[s] * norm ----------------
// 16 threads per edge, each moves one float4 (L2-resident gather + 4 fp32 atomics).
__global__ void k_scatter1(const int* __restrict__ src, const int* __restrict__ dst,
                           const float* __restrict__ dinv, const float* __restrict__ h1,
                           float* __restrict__ agg1, int e) {
    const long long tid   = (long long)blockIdx.x * blockDim.x + threadIdx.x;
    const long long total = (long long)e * 16;
    if (tid >= total) return;
    const int edge = (int)(tid >> 4);
    const int c4   = ((int)tid & 15) * 4;
    const int s = src[edge];
    const int d = dst[edge];
    const float norm = dinv[s] * dinv[d];
    const float4 hv = *(const float4*)(h1 + (long long)s * HID_DIM + c4);
    const float m0 = hv.x * norm, m1 = hv.y * norm, m2 = hv.z * norm, m3 = hv.w * norm;
    float* base = agg1 + (long long)d * HID_DIM + c4;
    atomicAdd(base + 0, m0);
    atomicAdd(base + 1, m1);
    atomicAdd(base + 2, m2);
    atomicAdd(base + 3, m3);
}

// ---------------- GEMM2: h2 = ReLU(agg1 + b1) @ W2 ; agg2 = h2 * dinv^2 ----------------
// Bias + ReLU fused into the A-operand load. One wave -> 16x16 output tile.
__global__ void k_gemm2(const float* __restrict__ agg1, const float* __restrict__ b1,
                        const float* __restrict__ W2, const float* __restrict__ dinv,
                        float* __restrict__ h2, float* __restrict__ agg2, int n) {
    const int lane  = threadIdx.x & 31;
    const int wave  = threadIdx.x >> 5;
    const int rtile = blockIdx.x * 8 + wave;
    const int rbase = rtile * 16;
    if (rbase >= n) return;

    const int half = lane >> 4;
    const int mrow = lane & 15;
    int arow = rbase + mrow;
    if (arow > n - 1) arow = n - 1;

    v8f acc;
#pragma unroll
    for (int q = 0; q < 8; ++q) acc[q] = 0.0f;

    for (int k0 = 0; k0 < HID_DIM; k0 += 4) {
        const int ka = k0 + 2 * half;
        v2f a;
        a.x = fmaxf(agg1[(long long)arow * HID_DIM + ka]     + b1[ka],     0.0f);
        a.y = fmaxf(agg1[(long long)arow * HID_DIM + ka + 1] + b1[ka + 1], 0.0f);
        v2f b;
        b.x = W2[ka * OUT_DIM + mrow];
        b.y = W2[(ka + 1) * OUT_DIM + mrow];
        acc = __builtin_amdgcn_wmma_f32_16x16x4_f32(
            false, a, false, b, (short)0, acc, false, false);
    }

    if (rbase + 16 <= n) {
#pragma unroll
        for (int g = 0; g < 8; ++g) {
            const int row = rbase + g + 8 * half;
            const float dv = dinv[row];
            const float sl = dv * dv;
            const float v = acc[g];
            h2[(long long)row * OUT_DIM + mrow]   = v;
            agg2[(long long)row * OUT_DIM + mrow] = v * sl;      // self-loop init
        }
    } else {
#pragma unroll
        for (int g = 0; g < 8; ++g) {
            const int row = rbase + g + 8 * half;
            if (row < n) {
                const float dv = dinv[row];
                const float sl = dv * dv;
                const float v = acc[g];
                h2[(long long)row * OUT_DIM + mrow]   = v;
                agg2[(long long)row * OUT_DIM + mrow] = v * sl;
            }
        }
    }
}

// ---------------- scatter layer 2: agg2[d] += h2[s] * norm ----------------
__global__ void k_scatter2(const int* __restrict__ src, const int* __restrict__ dst,
                           const float* __restrict__ dinv, const float* __restrict__ h2,
                           float* __restrict__ agg2, int e) {
    const long long tid   = (long long)blockIdx.x * blockDim.x + threadIdx.x;
    const long long total = (long long)e * 4;
    if (tid >= total) return;
    const int edge = (int)(tid >> 2);
    const int c4   = ((int)tid & 3) * 4;
    const int s = src[edge];
    const int d = dst[edge];
    const float norm = dinv[s] * dinv[d];
    const float4 hv = *(const float4*)(h2 + (long long)s * OUT_DIM + c4);
    const float m0 = hv.x * norm, m1 = hv.y * norm, m2 = hv.z * norm, m3 = hv.w * norm;
    float* base = agg2 + (long long)d * OUT_DIM + c4;
    atomicAdd(base + 0, m0);
    atomicAdd(base + 1, m1);
    atomicAdd(base + 2, m2);
    atomicAdd(base + 3, m3);
}

// ---------------- finalize: out = log_softmax(agg2 + b2) ----------------
__global__ void k_logsoftmax(const float* __restrict__ agg2, const float* __restrict__ b2,
                             float* __restrict__ out, int n) {
    const int i = blockIdx.x * blockDim.x + threadIdx.x;
    if (i >= n) return;
    float v[OUT_DIM];
    float m = -3.0e38f;
#pragma unroll
    for (int j = 0; j < OUT_DIM; ++j) {
        v[j] = agg2[(long long)i * OUT_DIM + j] + b2[j];
        m = fmaxf(m, v[j]);
    }
    float s = 0.0f;
#pragma unroll
    for (int j = 0; j < OUT_DIM; ++j) s += expf(v[j] - m);
    const float l = m + logf(s);
#pragma unroll
    for (int j = 0; j < OUT_DIM; ++j) out[(long long)i * OUT_DIM + j] = v[j] - l;
}

extern "C" void kernel_launch(void* const* d_in, const int* in_sizes, int n_in,
                              void* d_out, int out_size, void* d_ws, size_t ws_size,
                              hipStream_t stream) {
    const float* x  = (const float*)d_in[0];
    const int*   ei = (const int*)d_in[1];
    const float* W1 = (const float*)d_in[2];
    const float* b1 = (const float*)d_in[3];
    const float* W2 = (const float*)d_in[4];
    const float* b2 = (const float*)d_in[5];
    float* out = (float*)d_out;

    const int n = in_sizes[0] / IN_DIM;     // 100000
    const int e = in_sizes[1] / 2;          // 1600000
    const int* src = ei;                    // edge_index[0]
    const int* dst = ei + e;                // edge_index[1]

    // workspace partition (all fp32; N*(1+64+64+16+16)*4B ~= 64.4 MB, L2-resident)
    float* ws   = (float*)d_ws;
    float* dinv = ws;
    float* h1   = dinv + n;
    float* agg1 = h1   + (size_t)n * HID_DIM;
    float* h2   = agg1 + (size_t)n * HID_DIM;
    float* agg2 = h2   + (size_t)n * OUT_DIM;

    const int T = 256;
    k_deg_init<<<(n + T - 1) / T, T, 0, stream>>>(dinv, n);
    k_deg_count<<<(e + T - 1) / T, T, 0, stream>>>(dst, dinv, e);
    k_dinv<<<(n + T - 1) / T, T, 0, stream>>>(dinv, n);

    const int rowTiles = (n + 15) / 16;
    const int gemmBlocks = (rowTiles + 7) / 8;          // 8 waves/block, 1 tile/wave
    k_gemm1<<<gemmBlocks, 256, 0, stream>>>(x, W1, dinv, h1, agg1, n);

    const long long t1 = (long long)e * 16;
    k_scatter1<<<(int)((t1 + T - 1) / T), T, 0, stream>>>(src, dst, dinv, h1, agg1, e);

    k_gemm2<<<gemmBlocks, 256, 0, stream>>>(agg1, b1, W2, dinv, h2, agg2, n);

    const long long t2 = (long long)e * 4;
    k_scatter2<<<(int)((t2 + T - 1) / T), T, 0, stream>>>(src, dst, dinv, h2, agg2, e);

    k_logsoftmax<<<(n + T - 1) / T, T, 0, stream>>>(agg2, b2, out, n);
}